// DKSTE_85315230367936
// MI455X (gfx1250) — compile-verified
//
#include <hip/hip_runtime.h>
#include <hip/hip_bf16.h>

typedef float v2f __attribute__((ext_vector_type(2)));
typedef float v8f __attribute__((ext_vector_type(8)));

__device__ __forceinline__ float signf(float v) {
    // jnp.sign semantics: -1 / 0 / +1
    return (float)((v > 0.0f) - (v < 0.0f));
}

// One wave32 per batch element. 8 waves (256 threads) per block.
// entity_embedding: [NENTITY][512][1][2] f32 -> row of 1024 contiguous floats
// relation_embedding: [NRELATION][512][2]  -> row of 1024 contiguous floats
// alpha_embedding: [NRELATION][512]        -> row of 512 contiguous floats
__global__ __launch_bounds__(256) void
dkste_score_kernel(const int* __restrict__ head_idx,
                   const int* __restrict__ rel_idx,
                   const int* __restrict__ tail_idx,
                   const float* __restrict__ ent,
                   const float* __restrict__ rel,
                   const float* __restrict__ alp,
                   float* __restrict__ out,
                   int batch)
{
    const int wave = threadIdx.x >> 5;
    const int lane = threadIdx.x & 31;
    const int b = blockIdx.x * 8 + wave;
    if (b >= batch) return;   // wave-uniform branch: EXEC stays all-ones inside

    const long long h = head_idx[b];
    const long long r = rel_idx[b];
    const long long t = tail_idx[b];

    const float4* __restrict__ hp = (const float4*)(ent + h * 1024ll);
    const float4* __restrict__ tp = (const float4*)(ent + t * 1024ll);
    const float4* __restrict__ rp = (const float4*)(rel + r * 1024ll);
    const float2* __restrict__ ap = (const float2*)(alp + r * 512ll);

    // Each float4 j covers dims {2j, 2j+1} with (k=0,k=1) interleaved.
    // Lane L, iter i -> j = L + 32*i ; 8 iters * 32 lanes = 256 float4 = full row.
    float acc0 = 0.0f, acc1 = 0.0f;
#pragma unroll
    for (int i = 0; i < 8; ++i) {
        const int j = lane + 32 * i;
        const float4 hv = hp[j];
        const float4 tv = tp[j];
        const float4 rv = rp[j];
        const float2 av = ap[j];

        // dim 2j
        float x = signf(rv.x), y = signf(rv.y), a = signf(av.x);
        float s = 0.5f * (x + y), d = 0.5f * (x - y);
        float sc = hv.x * (s * tv.x - d * a * tv.y)
                 + hv.y * (d * tv.x + s * a * tv.y);
        float q = sc * sc;
        // dim 2j+1
        x = signf(rv.z); y = signf(rv.w); a = signf(av.y);
        s = 0.5f * (x + y); d = 0.5f * (x - y);
        sc = hv.z * (s * tv.z - d * a * tv.w)
           + hv.w * (d * tv.z + s * a * tv.w);
        q += sc * sc;

        if (i < 4) acc0 += q; else acc1 += q;
    }

    // Wave-wide sum of the 64 partials via V_WMMA_F32_16X16X4_F32.
    // A (16x4 f32, 2 VGPRs x 32 lanes) <- {acc0, acc1} per lane (bijective onto A).
    // B = all-ones (layout-independent). C = 0. Then D[m][n] = rowsum_m for every n,
    // so sum over ALL 256 D elements == 16 * (sum of all A elements), regardless of
    // the exact lane striping of A or D.
    v2f A; A[0] = acc0; A[1] = acc1;
    v2f B; B[0] = 1.0f; B[1] = 1.0f;
    v8f C = {};
    v8f D = __builtin_amdgcn_wmma_f32_16x16x4_f32(
        /*neg_a=*/false, A, /*neg_b=*/false, B,
        /*c_mod=*/(short)0, C, /*reuse_a=*/false, /*reuse_b=*/false);

    float lsum = D[0] + D[1] + D[2] + D[3] + D[4] + D[5] + D[6] + D[7];
#pragma unroll
    for (int m = 16; m >= 1; m >>= 1)
        lsum += __shfl_xor(lsum, m, 32);

    if (lane == 0)
        out[b] = sqrtf(lsum * (1.0f / 16.0f));
}

extern "C" void kernel_launch(void* const* d_in, const int* in_sizes, int n_in,
                              void* d_out, int out_size, void* d_ws, size_t ws_size,
                              hipStream_t stream) {
    const int*   head_idx = (const int*)d_in[0];
    const int*   rel_idx  = (const int*)d_in[1];
    const int*   tail_idx = (const int*)d_in[2];
    const float* ent      = (const float*)d_in[3];
    const float* rel      = (const float*)d_in[4];
    const float* alp      = (const float*)d_in[5];
    float*       out      = (float*)d_out;

    const int batch = in_sizes[0];           // 8192
    const int waves_per_block = 8;           // 256 threads
    const int grid = (batch + waves_per_block - 1) / waves_per_block;

    dkste_score_kernel<<<grid, 256, 0, stream>>>(head_idx, rel_idx, tail_idx,
                                                 ent, rel, alp, out, batch);
}